// NoveltyMap_31550829756579
// MI455X (gfx1250) — compile-verified
//
#include <hip/hip_runtime.h>
#include <hip/hip_bf16.h>

// ---------------- problem constants (from reference) ----------------
#define BATCH 4
#define CH    3
#define HH    256          // x is (4,3,256,256) fp32
#define KK    16
#define SS    2
#define PY    121          // (256-16)/2+1
#define PXN   121
#define HY    1936         // y is (4,3,1936,1936) fp32
#define GXN   8            // ceil(121/16) patch-groups per row

typedef float v2f __attribute__((ext_vector_type(2)));
typedef float v8f __attribute__((ext_vector_type(8)));

// Stage 1: one wave32 per (b, py, 16-patch group). Each lane streams two of the
// four interleaved k-streams of one patch (768 elems total/patch) with b64
// loads + v_fmac_f32. The 4 fp32 partials per patch form a 16x4 f32 A-matrix;
// a single V_WMMA_F32_16X16X4_F32 against an all-ones B reduces them to the
// per-patch sum in full fp32 (matrix unit as cross-lane reducer).
__global__ __launch_bounds__(256) void novelty_stage1(
    const float* __restrict__ x, const float* __restrict__ y,
    float* __restrict__ m)
{
    const int wave = blockIdx.x * (blockDim.x >> 5) + (threadIdx.x >> 5);
    const int lane = threadIdx.x & 31;

    // decompose wave -> (b, py, group)
    int b   = wave / (PY * GXN);
    int r   = wave - b * (PY * GXN);
    int py  = r / GXN;
    int gx  = r - py * GXN;
    int px0 = gx * 16;

    const int p    = lane & 15;   // patch within group
    const int half = lane >> 4;   // which pair of k-streams (K=0,1 vs K=2,3)
    int px = px0 + p;
    if (px > PXN - 1) px = PXN - 1;   // clamp loads for tail group (EXEC stays full)

    const float* xb = x + (size_t)b * CH * HH * HH + (size_t)(SS * py) * HH + SS * px;
    const float* yb = y + (size_t)b * CH * HY * HY + (size_t)(KK * py) * HY + KK * px;

    float acc0 = 0.0f, acc1 = 0.0f;
    for (int c = 0; c < CH; ++c) {
        const float* xc = xb + (size_t)c * HH * HH;
        const float* yc = yb + (size_t)c * HY * HY;
        for (int ky = 0; ky < KK; ++ky) {
            const float* xr = xc + ky * HH;
            const float* yr = yc + ky * HY;
            #pragma unroll
            for (int kxb = 0; kxb < KK; kxb += 4) {
                const int kx = kxb + 2 * half;          // 8B-aligned (even index)
                const float2 xv = *(const float2*)(xr + kx);
                const float2 yv = *(const float2*)(yr + kx);
                float d0 = xv.x - yv.x;
                float d1 = xv.y - yv.y;
                acc0 = fmaf(d0, d0, acc0);
                acc1 = fmaf(d1, d1, acc1);
            }
        }
    }

    // A[16x4] f32: lane p holds (K=0,K=1), lane p+16 holds (K=2,K=3).
    v2f A;  A.x = acc0; A.y = acc1;
    v2f Bm; Bm.x = 1.0f; Bm.y = 1.0f;     // B = ones(4x16): D[p][n] = sum_k A[p][k]
    v8f Cm = {};
    Cm = __builtin_amdgcn_wmma_f32_16x16x4_f32(
        /*neg_a=*/false, A, /*neg_b=*/false, Bm,
        /*c_mod=*/(short)0, Cm, /*reuse_a=*/false, /*reuse_b=*/false);

    // Column N=0 of D lives in lane 0 (M=0..7) and lane 16 (M=8..15).
    if ((lane & 15) == 0) {
        const int   mrow = half * 8;
        float*      mout = m + ((size_t)b * PY + py) * PXN + px0;
        const float sc   = 1.0f / 768.0f;  // mean over C*K*K
        #pragma unroll
        for (int i = 0; i < 8; ++i) {
            const int pp = mrow + i;
            if (px0 + pp < PXN) mout[pp] = Cm[i] * sc;
        }
    }
}

// Stage 2: bilinear 121x121 -> 256x256, jax half-pixel convention.
// Edge weight renormalization in jax == index clamping for the linear kernel.
__global__ __launch_bounds__(256) void novelty_resize(
    const float* __restrict__ m, float* __restrict__ out)
{
    int idx = blockIdx.x * blockDim.x + threadIdx.x;
    if (idx >= BATCH * HH * HH) return;
    int b = idx >> 16;
    int r = idx & 65535;
    int i = r >> 8;
    int j = r & 255;

    const float scale = (float)PY / (float)HH;   // 121/256
    float fy = (i + 0.5f) * scale - 0.5f;
    float fx = (j + 0.5f) * scale - 0.5f;
    float yf = floorf(fy), xf = floorf(fx);
    int   y0 = (int)yf,   x0 = (int)xf;
    float wy = fy - yf,   wx = fx - xf;

    int y0c = min(max(y0,     0), PY  - 1);
    int y1c = min(max(y0 + 1, 0), PY  - 1);
    int x0c = min(max(x0,     0), PXN - 1);
    int x1c = min(max(x0 + 1, 0), PXN - 1);

    const float* mb = m + (size_t)b * PY * PXN;
    float v00 = mb[y0c * PXN + x0c];
    float v01 = mb[y0c * PXN + x1c];
    float v10 = mb[y1c * PXN + x0c];
    float v11 = mb[y1c * PXN + x1c];
    float v0 = v00 + wx * (v01 - v00);
    float v1 = v10 + wx * (v11 - v10);
    out[idx] = v0 + wy * (v1 - v0);
}

extern "C" void kernel_launch(void* const* d_in, const int* in_sizes, int n_in,
                              void* d_out, int out_size, void* d_ws, size_t ws_size,
                              hipStream_t stream) {
    const float* x = (const float*)d_in[0];   // (4,3,256,256) fp32
    const float* y = (const float*)d_in[1];   // (4,3,1936,1936) fp32
    float* out = (float*)d_out;               // (4,256,256) fp32
    float* m   = (float*)d_ws;                // 4*121*121 floats scratch

    // 4*121*8 = 3872 waves, 8 waves/block -> 484 blocks
    novelty_stage1<<<(BATCH * PY * GXN) / 8, 256, 0, stream>>>(x, y, m);

    const int n_out = BATCH * HH * HH;
    novelty_resize<<<(n_out + 255) / 256, 256, 0, stream>>>(m, out);
}